// LogSlaterDeterminant_65532611002542
// MI455X (gfx1250) — compile-verified
//
#include <hip/hip_runtime.h>
#include <math.h>

#define NORB 256
#define NF   128
#define LDA  132      // pad to de-conflict column accesses (132 % 64 == 4)
#define NB   16
#define NTHREADS 256  // 8 waves of 32

typedef __attribute__((ext_vector_type(2))) float v2f;
typedef __attribute__((ext_vector_type(8))) float v8f;

__global__ __launch_bounds__(NTHREADS) void slogdet_lu_wmma_kernel(
    const int* __restrict__ nOcc, const float* __restrict__ M, float* __restrict__ out)
{
    __shared__ float As[NF][LDA];
    __shared__ int   scanBuf[NTHREADS];
    __shared__ int   occ[NF];
    __shared__ float redV[NF];
    __shared__ int   redI[NF];
    __shared__ int   pivParity;

    const int tid  = threadIdx.x;
    const int b    = blockIdx.x;
    const int lane = tid & 31;
    const int waveId = tid >> 5;

    // --- 1. occupied-index extraction: inclusive scan over 256 orbitals ---
    int occBit = (nOcc[b * NORB + tid] > 0) ? 1 : 0;
    scanBuf[tid] = occBit;
    __syncthreads();
    for (int off = 1; off < NTHREADS; off <<= 1) {
        int add = (tid >= off) ? scanBuf[tid - off] : 0;
        __syncthreads();
        scanBuf[tid] += add;
        __syncthreads();
    }
    if (occBit) {
        int r = scanBuf[tid] - 1;          // exclusive rank = ascending position
        if (r < NF) occ[r] = tid;
    }
    if (tid == 0) pivParity = 0;
    __syncthreads();

    // --- 2. gather Slater matrix rows of M into LDS (coalesced on columns) ---
    for (int idx = tid; idx < NF * NF; idx += NTHREADS) {
        int r = idx >> 7;
        int c = idx & (NF - 1);
        As[r][c] = M[occ[r] * NF + c];
    }
    __syncthreads();

    // --- 3. blocked right-looking LU with partial pivoting ---
    // NOTE: L (below-diagonal) is stored NEGATED in LDS so the WMMA trailing
    // update needs no per-element negation (f32 WMMA has no A/B neg modifier).
    for (int p = 0; p < NF; p += NB) {
        // ---- panel factorization: columns p .. p+NB-1 ----
        for (int k = p; k < p + NB; ++k) {
            // pivot search over rows k..NF-1 (on raw updated entries)
            if (tid < NF) {
                int i = k + tid;
                if (i < NF) { redV[tid] = fabsf(As[i][k]); redI[tid] = i; }
                else        { redV[tid] = -1.0f;           redI[tid] = k; }
            }
            __syncthreads();
            for (int s = 64; s > 0; s >>= 1) {
                if (tid < s) {
                    if (redV[tid + s] > redV[tid]) {
                        redV[tid] = redV[tid + s];
                        redI[tid] = redI[tid + s];
                    }
                }
                __syncthreads();
            }
            int ip = redI[0];
            // full row swap (LAPACK style; negated-L entries swap consistently)
            if (ip != k) {
                if (tid < NF) {
                    float t0 = As[k][tid];
                    As[k][tid]  = As[ip][tid];
                    As[ip][tid] = t0;
                }
                if (tid == 0) pivParity ^= 1;
            }
            __syncthreads();
            float nrinv = -1.0f / As[k][k];
            // store negated multipliers: L[i][k] = -A[i][k]/U[k][k]
            for (int i = k + 1 + tid; i < NF; i += NTHREADS) As[i][k] *= nrinv;
            __syncthreads();
            // rank-1 update inside panel: A += (negL) * U_row  ==  A -= L*U
            int w = p + NB - 1 - k;
            if (w > 0) {
                int h = NF - 1 - k;
                int total = w * h;
                for (int idx = tid; idx < total; idx += NTHREADS) {
                    int i = k + 1 + idx / w;
                    int c = k + 1 + idx % w;
                    As[i][c] += As[i][k] * As[k][c];
                }
            }
            __syncthreads();
        }

        int cb = p + NB;                    // first trailing column
        if (cb >= NF) break;
        int wt = NF - cb;                   // trailing width

        // ---- TRSM: U12 = L11^{-1} * A12 (forward substitution with negated L) ----
        for (int k = p; k < p + NB - 1; ++k) {
            int h = p + NB - 1 - k;
            int total = h * wt;
            for (int idx = tid; idx < total; idx += NTHREADS) {
                int i = k + 1 + idx / wt;
                int c = cb + idx % wt;
                As[i][c] += As[i][k] * As[k][c];
            }
            __syncthreads();
        }

        // ---- GEMM trailing update: A22 += (negL21) * U12 via v_wmma_f32_16x16x4_f32 ----
        int nt = wt / NB;                   // 16x16 tiles per dimension
        int nTiles = nt * nt;
        int nLo = lane & 15;                // N (column) index within tile
        int hi  = lane >> 4;                // half-wave selector
        for (int t = waveId; t < nTiles; t += (NTHREADS / 32)) {
            int R   = cb + (t / nt) * NB;   // tile row base
            int CB2 = cb + (t % nt) * NB;   // tile col base
            // C/D fragment: VGPR v holds C[v + 8*hi][nLo]
            v8f acc;
            #pragma unroll
            for (int vv = 0; vv < 8; ++vv) acc[vv] = As[R + vv + 8 * hi][CB2 + nLo];
            // K = 16 as 4 chained 16x16x4 WMMAs
            #pragma unroll
            for (int kk = 0; kk < 4; ++kk) {
                int kb = p + 4 * kk + 2 * hi;
                v2f a, bf;
                // A fragment = negated L (already negated in LDS)
                a[0] = As[R + nLo][kb];
                a[1] = As[R + nLo][kb + 1];
                // B fragment = U12 rows
                bf[0] = As[kb][CB2 + nLo];
                bf[1] = As[kb + 1][CB2 + nLo];
                acc = __builtin_amdgcn_wmma_f32_16x16x4_f32(
                    /*neg_a=*/false, a, /*neg_b=*/false, bf,
                    /*c_mod=*/(short)0, acc, /*reuse_a=*/false, /*reuse_b=*/false);
            }
            #pragma unroll
            for (int vv = 0; vv < 8; ++vv) As[R + vv + 8 * hi][CB2 + nLo] = acc[vv];
        }
        __syncthreads();
    }

    // --- 4. slogdet from U diagonal + pivot parity ---
    if (tid < NF) {
        float d = As[tid][tid];
        redV[tid] = logf(fabsf(d));
        redI[tid] = (d < 0.0f) ? 1 : 0;
    }
    __syncthreads();
    for (int s = 64; s > 0; s >>= 1) {
        if (tid < s) { redV[tid] += redV[tid + s]; redI[tid] += redI[tid + s]; }
        __syncthreads();
    }
    if (tid == 0) {
        int neg = (redI[0] + pivParity) & 1;
        out[2 * b]     = redV[0];                                    // Re: log|det|
        out[2 * b + 1] = neg ? 3.14159265358979323846f : 0.0f;       // Im: log(sign)
    }
}

extern "C" void kernel_launch(void* const* d_in, const int* in_sizes, int n_in,
                              void* d_out, int out_size, void* d_ws, size_t ws_size,
                              hipStream_t stream) {
    const int*   n = (const int*)d_in[0];    // [B, 256] int32 occupations
    const float* M = (const float*)d_in[1];  // [256, 128] f32 orbital matrix
    float* out = (float*)d_out;              // complex64 -> interleaved (re, im)
    int batch = in_sizes[0] / NORB;
    slogdet_lu_wmma_kernel<<<batch, NTHREADS, 0, stream>>>(n, M, out);
}